// DeltaRuleGated_22488448762149
// MI455X (gfx1250) — compile-verified
//
#include <hip/hip_runtime.h>

// Gated delta-rule scan for MI455X (gfx1250, wave32).
// B=4, T=2048, H=8, D=128. Columns of the 128x128 state are independent
// recurrences -> parallelize over (b, h, column-group). State lives in VGPRs
// (packed f32 math). Per-step inputs (q,k,v,f,g) are double-buffered into LDS
// with the CDNA5 async global->LDS engine (ASYNCcnt), synced per 8-step chunk.
// The K-reduction of the q.M dot products runs on the matrix pipe via
// V_WMMA_F32_16X16X4_F32 (A = ones -> D[m,n] = po[n] + po[n+16]); the final
// 16->1 sum uses DPP row_xmask adds (no LDS). Kernel is divergence-free.

#define B_ 4
#define T_ 2048
#define H_ 8
#define D_ 128
#define FMIN_ 0.8f

#define CHUNK 8                    // time steps staged per LDS buffer
#define NCHUNK (T_ / CHUNK)        // 256
#define THREADS 128                // 4 waves of 32
#define COLS_PER_WG 16
#define COLG (D_ / COLS_PER_WG)    // 8 column groups -> grid.x
#define COLS_PER_WAVE 4

#define ARR_STRIDE (CHUNK * D_)    // floats per array per buffer (1024)
#define BUF_STRIDE (5 * ARR_STRIDE)

#if defined(__has_builtin)
#  if __has_builtin(__builtin_amdgcn_global_load_async_to_lds_b128)
#    define HAVE_ASYNC_LDS 1
#  endif
#  if __has_builtin(__builtin_amdgcn_s_wait_asynccnt)
#    define HAVE_WAIT_ASYNC 1
#  endif
#  if __has_builtin(__builtin_amdgcn_wmma_f32_16x16x4_f32)
#    define HAVE_WMMA_F32X4 1
#  endif
#endif

typedef int   v4i __attribute__((ext_vector_type(4)));
typedef float v2f __attribute__((ext_vector_type(2)));
typedef float v8f __attribute__((ext_vector_type(8)));
typedef __attribute__((address_space(1))) v4i GlobV4;   // global
typedef __attribute__((address_space(3))) v4i LdsV4;    // LDS

__device__ __forceinline__ void async_cp16(float* lds_dst, const float* gsrc) {
#if defined(HAVE_ASYNC_LDS)
  // async copy 16B global -> LDS, tracked by ASYNCcnt (CDNA5 path)
  __builtin_amdgcn_global_load_async_to_lds_b128(
      (GlobV4*)(gsrc), (LdsV4*)(lds_dst), 0, 0);
#else
  *(float4*)lds_dst = *(const float4*)gsrc;
#endif
}

__device__ __forceinline__ void chunk_sync() {
#if defined(HAVE_ASYNC_LDS) && defined(HAVE_WAIT_ASYNC)
  __builtin_amdgcn_s_wait_asynccnt(0);
#endif
  __syncthreads();
}

// x + (x from lane ^ XORMASK within its row of 16), via DPP row_xmask.
template <int XORMASK>
__device__ __forceinline__ float dpp_xor_add(float x) {
  const int xi = __builtin_bit_cast(int, x);
  const int ti = __builtin_amdgcn_mov_dpp(xi, 0x160 | XORMASK, 0xf, 0xf, true);
  return x + __builtin_bit_cast(float, ti);
}

// Reduce a per-lane partial (32 lanes) to the wave-wide sum in all lanes.
__device__ __forceinline__ float wave_reduce(float po) {
#if defined(HAVE_WMMA_F32X4)
  // Matrix-pipe 32->16 reduction: A = ones(16x4), B = (po, 0) so that
  // D[m,n] = sum_k B[k,n] = po[n] + po[n+16], independent of B row order.
  const v2f a = {1.0f, 1.0f};
  const v2f bcol = {po, 0.0f};
  v8f c = {};
  c = __builtin_amdgcn_wmma_f32_16x16x4_f32(false, a, false, bcol,
                                            (short)0, c, false, false);
  float r = c[0];                 // r[lane] = red16[lane & 15]
  r = dpp_xor_add<8>(r);          // 16 -> 1 within each row of 16 (VALU only)
  r = dpp_xor_add<4>(r);
  r = dpp_xor_add<2>(r);
  r = dpp_xor_add<1>(r);
  return r;
#else
  po += __shfl_xor(po, 16, 32);
  po += __shfl_xor(po,  8, 32);
  po += __shfl_xor(po,  4, 32);
  po += __shfl_xor(po,  2, 32);
  po += __shfl_xor(po,  1, 32);
  return po;
#endif
}

// Stage CHUNK steps of all five per-head vectors into one LDS buffer.
__device__ __forceinline__ void stage_chunk(float* buf,
                                            const float* __restrict__ qh,
                                            const float* __restrict__ kh,
                                            const float* __restrict__ vh,
                                            const float* __restrict__ fh,
                                            const float* __restrict__ gh,
                                            int t0, int tid) {
  const size_t step_stride = (size_t)H_ * D_;
#pragma unroll
  for (int a = 0; a < 5; ++a) {
    const float* base = (a == 0) ? qh : (a == 1) ? kh : (a == 2) ? vh
                                      : (a == 3) ? fh : gh;
    float* abuf = buf + a * ARR_STRIDE;
#pragma unroll
    for (int i = tid; i < CHUNK * (D_ / 4); i += THREADS) {  // 256 xfers, 2/thread
      const int s  = i >> 5;          // D_/4 == 32 per step
      const int d4 = (i & 31) << 2;
      async_cp16(abuf + s * D_ + d4,
                 base + (size_t)(t0 + s) * step_stride + d4);
    }
  }
}

__global__ __launch_bounds__(THREADS)
void DeltaRuleGated_kernel(const float* __restrict__ q,
                           const float* __restrict__ k,
                           const float* __restrict__ v,
                           const float* __restrict__ f,
                           const float* __restrict__ g,
                           float* __restrict__ out) {
  __shared__ __align__(16) float lds[2 * BUF_STRIDE];  // 40 KB

  const int cg   = blockIdx.x;          // column group (0..7)
  const int h    = blockIdx.y;          // head
  const int b    = blockIdx.z;          // batch
  const int tid  = threadIdx.x;
  const int wave = tid >> 5;
  const int lane = tid & 31;
  const int cbase = cg * COLS_PER_WG + wave * COLS_PER_WAVE;  // multiple of 4
  const int cl   = lane & 3;            // column captured by this lane
  const int sl   = lane >> 2;           // step captured by this lane (0..7)

  const size_t headbase = ((size_t)b * T_ * H_ + h) * D_;
  const float* qh = q + headbase;
  const float* kh = k + headbase;
  const float* vh = v + headbase;
  const float* fh = f + headbase;
  const float* gh = g + headbase;

  // Each lane owns output element (t = ci*CHUNK + sl, e = cbase + cl):
  // one coalesced, non-predicated store per chunk.
  float* optr = out + headbase + (size_t)sl * (H_ * D_) + cbase + cl;

  // State: 4 columns x 128 rows, rows striped 4-per-lane -> 16 VGPRs/lane.
  float m[COLS_PER_WAVE][4];
#pragma unroll
  for (int c = 0; c < COLS_PER_WAVE; ++c)
#pragma unroll
    for (int r = 0; r < 4; ++r) m[c][r] = 0.0f;

  stage_chunk(lds, qh, kh, vh, fh, gh, 0, tid);
  chunk_sync();

  for (int ci = 0; ci < NCHUNK; ++ci) {
    float* cur = lds + (ci & 1) * BUF_STRIDE;
    if (ci + 1 < NCHUNK)  // prefetch next chunk into the other buffer (async)
      stage_chunk(lds + ((ci + 1) & 1) * BUF_STRIDE, qh, kh, vh, fh, gh,
                  (ci + 1) * CHUNK, tid);

    const float* Q = cur + 0 * ARR_STRIDE;
    const float* K = cur + 1 * ARR_STRIDE;
    const float* V = cur + 2 * ARR_STRIDE;
    const float* F = cur + 3 * ARR_STRIDE;
    const float* G = cur + 4 * ARR_STRIDE;

    float oreg = 0.0f;  // this lane's output for (sl, cl) of this chunk

#pragma unroll
    for (int s = 0; s < CHUNK; ++s) {
      // d-side vectors: d = 4*lane + r (conflict-free ds_load_b128)
      const float4 qv = *(const float4*)(Q + s * D_ + 4 * lane);
      const float4 kv = *(const float4*)(K + s * D_ + 4 * lane);
      const float4 gv = *(const float4*)(G + s * D_ + 4 * lane);
      const float4 fv = *(const float4*)(F + s * D_ + 4 * lane);
      const float kg0 = kv.x * gv.x, kg1 = kv.y * gv.y;
      const float kg2 = kv.z * gv.z, kg3 = kv.w * gv.w;

      // column-side scalars: 3 uniform ds_load_b128 for the wave's 4 columns
      const float4 fc4 = *(const float4*)(F + s * D_ + cbase);
      const float4 vc4 = *(const float4*)(V + s * D_ + cbase);
      const float4 gc4 = *(const float4*)(G + s * D_ + cbase);
      const float fe_[4]  = {fc4.x, fc4.y, fc4.z, fc4.w};
      const float vge_[4] = {vc4.x * gc4.x, vc4.y * gc4.y,
                             vc4.z * gc4.z, vc4.w * gc4.w};

      float osum[COLS_PER_WAVE];
#pragma unroll
      for (int c = 0; c < COLS_PER_WAVE; ++c) {
        const float fe  = fe_[c];
        const float vge = vge_[c];
        float po;
        {
          const float fo = __builtin_amdgcn_fmed3f(fv.x * fe, FMIN_, 1.0f);
          m[c][0] = m[c][0] * fo + kg0 * vge;  po  = qv.x * m[c][0];
        }
        {
          const float fo = __builtin_amdgcn_fmed3f(fv.y * fe, FMIN_, 1.0f);
          m[c][1] = m[c][1] * fo + kg1 * vge;  po += qv.y * m[c][1];
        }
        {
          const float fo = __builtin_amdgcn_fmed3f(fv.z * fe, FMIN_, 1.0f);
          m[c][2] = m[c][2] * fo + kg2 * vge;  po += qv.z * m[c][2];
        }
        {
          const float fo = __builtin_amdgcn_fmed3f(fv.w * fe, FMIN_, 1.0f);
          m[c][3] = m[c][3] * fo + kg3 * vge;  po += qv.w * m[c][3];
        }
        osum[c] = wave_reduce(po);   // WMMA 32->16 + 4 DPP row_xmask adds
      }

      // Branch-free capture: this lane keeps (sl, cl)'s value.
      float pick = osum[0];
      pick = (cl == 1) ? osum[1] : pick;
      pick = (cl == 2) ? osum[2] : pick;
      pick = (cl == 3) ? osum[3] : pick;
      oreg = (sl == s) ? pick : oreg;
    }

    *optr = oreg;                       // one coalesced store per chunk
    optr += (size_t)CHUNK * (H_ * D_);

    chunk_sync();  // wait our async prefetch + all waves done with buffers
  }
}

extern "C" void kernel_launch(void* const* d_in, const int* in_sizes, int n_in,
                              void* d_out, int out_size, void* d_ws, size_t ws_size,
                              hipStream_t stream) {
  (void)in_sizes; (void)n_in; (void)d_ws; (void)ws_size; (void)out_size;
  const float* q = (const float*)d_in[0];
  const float* k = (const float*)d_in[1];
  const float* v = (const float*)d_in[2];
  const float* f = (const float*)d_in[3];
  const float* g = (const float*)d_in[4];
  float* out = (float*)d_out;

  dim3 grid(COLG, H_, B_);   // 8 x 8 x 4 = 256 workgroups
  dim3 block(THREADS);       // 4 waves of 32
  hipLaunchKernelGGL(DeltaRuleGated_kernel, grid, block, 0, stream,
                     q, k, v, f, g, out);
}